// GlobalAttention_5368709120302
// MI455X (gfx1250) — compile-verified
//
#include <hip/hip_runtime.h>
#include <math.h>

// Problem constants (match reference)
#define B_  256
#define N_  2048
#define C_  64
#define H_  8
#define KD_ 8
#define VD_ 8
#define OD_ 64
#define HV_ 64   // H*VD

typedef __attribute__((ext_vector_type(2))) float v2f;
typedef __attribute__((ext_vector_type(8))) float v8f;

// ---------------------------------------------------------------------------
// Kernel A: per-batch pooled query + streaming online-softmax attention.
// grid = (B), block = 256 threads (8 wave32). Output: wa[B][64] in workspace.
// ---------------------------------------------------------------------------
__global__ __launch_bounds__(256) void ga_pool_attn(
    const float* __restrict__ qd,   // [B,N,C]
    const float* __restrict__ md,   // [B,N,C]
    const float* __restrict__ qm,   // [B,N,C] (broadcast mask; use [...,0])
    const float* __restrict__ qw,   // [C,H,KD] = [64,64]
    const float* __restrict__ kw,   // [C,KD]   = [64,8]
    const float* __restrict__ vw,   // [C,VD]   = [64,8]
    float* __restrict__ wa)         // [B,64]
{
  __shared__ float s_kw[C_][KD_];
  __shared__ float s_vw[C_][VD_];
  __shared__ float s_red[4][C_];
  __shared__ float s_msum[4];
  __shared__ float s_qavg[C_];
  __shared__ float s_q[H_][KD_];
  __shared__ float s_mrg[8][H_][10];   // per-wave partials: m, s, acc[8]

  const int b   = blockIdx.x;
  const int tid = threadIdx.x;
  const size_t rowbase = (size_t)b * N_;

  // stage k/v projection weights into LDS
  for (int i = tid; i < C_ * KD_; i += 256) {
    s_kw[i >> 3][i & 7] = kw[i];
    s_vw[i >> 3][i & 7] = vw[i];
  }

  // ---- phase 1: masked mean over N (q_avg) ----
  {
    const int c = tid & 63;
    const int g = tid >> 6;           // 0..3
    float qs = 0.f, ms = 0.f;
    for (int n = g; n < N_; n += 4) {
      const float mask = qm[(rowbase + n) * C_];      // broadcast channel 0
      const float x    = qd[(rowbase + n) * C_ + c];
      qs += mask * x;
      ms += mask;
    }
    s_red[g][c] = qs;
    if (c == 0) s_msum[g] = ms;
  }
  __syncthreads();
  if (tid < C_) {
    const float num = s_red[0][tid] + s_red[1][tid] + s_red[2][tid] + s_red[3][tid];
    const float den = s_msum[0] + s_msum[1] + s_msum[2] + s_msum[3];
    s_qavg[tid] = num / (den + 1e-10f);
  }
  __syncthreads();

  // ---- phase 2: pooled query projection q[h][kd], scaled 1/sqrt(KD) ----
  if (tid < HV_) {
    float acc = 0.f;
    for (int c = 0; c < C_; ++c) acc += s_qavg[c] * qw[c * 64 + tid];
    s_q[tid >> 3][tid & 7] = acc * 0.35355339059327373f;   // 8^-0.5
  }
  __syncthreads();

  // ---- phase 3: streaming k/v projection + online softmax + weighted avg ----
  float mh[H_], sh[H_], acc[H_][VD_];
  for (int h = 0; h < H_; ++h) {
    mh[h] = -1e30f; sh[h] = 0.f;
    for (int v = 0; v < VD_; ++v) acc[h][v] = 0.f;
  }

  for (int n = tid; n < N_; n += 256) {
    const float* mrow = md + (rowbase + n) * C_;
    const float  mask = qm[(rowbase + n) * C_];
    float kv[KD_] = {0,0,0,0,0,0,0,0};
    float vv[VD_] = {0,0,0,0,0,0,0,0};
    const float4* m4p = reinterpret_cast<const float4*>(mrow);
    for (int c4 = 0; c4 < C_ / 4; ++c4) {
      const float4 m4 = m4p[c4];
      const float xs[4] = {m4.x, m4.y, m4.z, m4.w};
      for (int j = 0; j < 4; ++j) {
        const float x = xs[j];
        const int   c = c4 * 4 + j;
        for (int d = 0; d < 8; ++d) {
          kv[d] += x * s_kw[c][d];
          vv[d] += x * s_vw[c][d];
        }
      }
    }
    const float mb = 1e9f * (mask - 1.0f);
    for (int h = 0; h < H_; ++h) {
      float lg = mb;
      for (int d = 0; d < KD_; ++d) lg += s_q[h][d] * kv[d];
      const float mnew = fmaxf(mh[h], lg);
      const float e1 = __expf(mh[h] - mnew);
      const float p  = __expf(lg - mnew);
      sh[h] = sh[h] * e1 + p;
      for (int v = 0; v < VD_; ++v) acc[h][v] = acc[h][v] * e1 + p * vv[v];
      mh[h] = mnew;
    }
  }

  // wave32 tree merge via shuffles
  for (int off = 16; off > 0; off >>= 1) {
    for (int h = 0; h < H_; ++h) {
      const float om = __shfl_xor(mh[h], off, 32);
      const float os = __shfl_xor(sh[h], off, 32);
      const float M  = fmaxf(mh[h], om);
      const float e1 = __expf(mh[h] - M);
      const float e2 = __expf(om   - M);
      float ns = sh[h] * e1 + os * e2;
      for (int v = 0; v < VD_; ++v) {
        const float oa = __shfl_xor(acc[h][v], off, 32);
        acc[h][v] = acc[h][v] * e1 + oa * e2;
      }
      sh[h] = ns; mh[h] = M;
    }
  }
  const int lane = tid & 31, wv = tid >> 5;
  if (lane == 0) {
    for (int h = 0; h < H_; ++h) {
      s_mrg[wv][h][0] = mh[h];
      s_mrg[wv][h][1] = sh[h];
      for (int v = 0; v < VD_; ++v) s_mrg[wv][h][2 + v] = acc[h][v];
    }
  }
  __syncthreads();
  if (tid < H_) {
    const int h = tid;
    float M = -1e30f;
    for (int w = 0; w < 8; ++w) M = fmaxf(M, s_mrg[w][h][0]);
    float s = 0.f, av[VD_] = {0,0,0,0,0,0,0,0};
    for (int w = 0; w < 8; ++w) {
      const float e = __expf(s_mrg[w][h][0] - M);
      s += e * s_mrg[w][h][1];
      for (int v = 0; v < VD_; ++v) av[v] += e * s_mrg[w][h][2 + v];
    }
    const float inv = 1.0f / s;
    for (int v = 0; v < VD_; ++v) wa[b * HV_ + h * VD_ + v] = av[v] * inv;
  }
}

// ---------------------------------------------------------------------------
// Kernel B: gate = sigmoid(X @ Gw + gb);  out = gate @ (wa ⊙ Ow) + ob
// grid = (B, N/64), block = 128 (4 wave32); each wave owns a 16-row panel.
// All matmuls via v_wmma_f32_16x16x4_f32 (full f32 — memory-bound problem).
// ---------------------------------------------------------------------------
__global__ __launch_bounds__(128) void ga_gate_out(
    const float* __restrict__ qd,   // [B,N,C]
    const float* __restrict__ gw,   // [C, H*VD] = [64,64]
    const float* __restrict__ gb,   // [H*VD]
    const float* __restrict__ ow,   // [H*VD, OD] = [64,64]
    const float* __restrict__ ob,   // [OD]
    const float* __restrict__ wa,   // [B, H*VD]
    float* __restrict__ out)        // [B,N,OD]
{
  __shared__ float sGw[C_ * HV_];     // 16 KB
  __shared__ float sWb[HV_ * OD_];    // 16 KB (wa folded into Ow)
  __shared__ float sX[4][16 * C_];    // 16 KB (per-wave X tile, reused for gate)
  __shared__ float sGb[HV_];
  __shared__ float sOb[OD_];

  const int b   = blockIdx.x;
  const int nt  = blockIdx.y;
  const int tid = threadIdx.x;

  for (int i = tid; i < C_ * HV_; i += 128) sGw[i] = gw[i];
  for (int i = tid; i < HV_ * OD_; i += 128) sWb[i] = ow[i] * wa[b * HV_ + (i >> 6)];
  if (tid < 64) { sGb[tid] = gb[tid]; sOb[tid] = ob[tid]; }

  const int wv   = tid >> 5;
  const int lane = tid & 31;
  const int half = lane >> 4;       // 0: K+{0,1}, 1: K+{2,3} (A-frag layout)
  const int l16  = lane & 15;
  const int rowbase = nt * 64 + wv * 16;
  const float* Xg = qd + ((size_t)b * N_ + rowbase) * C_;
  float* sx = sX[wv];

  // coalesced 16x64 f32 tile load into LDS (8 x float4 per lane)
  for (int i = 0; i < 8; ++i) {
    const int idx = lane * 4 + i * 128;
    *reinterpret_cast<float4*>(sx + idx) =
        *reinterpret_cast<const float4*>(Xg + idx);
  }
  if (nt + 1 < N_ / 64) __builtin_prefetch(Xg + 64 * C_, 0, 1);
  __syncthreads();

  // A fragments for this wave's 16 rows: 16 K-steps of 4
  v2f afrag[16];
  for (int kk = 0; kk < 16; ++kk) {
    const int k0 = kk * 4 + half * 2;
    afrag[kk].x = sx[l16 * C_ + k0];
    afrag[kk].y = sx[l16 * C_ + k0 + 1];
  }

  // ---- GEMM 1: gate logits (16x64 = 16x64 @ 64x64) ----
  v8f cts[4];
  for (int ct = 0; ct < 4; ++ct) {
    v8f c = {};
    for (int kk = 0; kk < 16; ++kk) {
      const int k0 = kk * 4 + half * 2;
      v2f bf;
      bf.x = sGw[(k0    ) * HV_ + ct * 16 + l16];
      bf.y = sGw[(k0 + 1) * HV_ + ct * 16 + l16];
      c = __builtin_amdgcn_wmma_f32_16x16x4_f32(false, afrag[kk], false, bf,
                                                (short)0, c, false, false);
    }
    cts[ct] = c;
  }
  __syncthreads();

  // sigmoid gate -> LDS (C-layout -> row-major, so it can be re-read as A)
  for (int ct = 0; ct < 4; ++ct) {
    const int nn = ct * 16 + l16;
    const float bias = sGb[nn];
    for (int j = 0; j < 8; ++j) {
      const int m = j + 8 * half;
      const float x = cts[ct][j] + bias;
      sx[m * C_ + nn] = 1.0f / (1.0f + __expf(-x));
    }
  }
  __syncthreads();

  // reload A fragments (now the gate tile)
  for (int kk = 0; kk < 16; ++kk) {
    const int k0 = kk * 4 + half * 2;
    afrag[kk].x = sx[l16 * C_ + k0];
    afrag[kk].y = sx[l16 * C_ + k0 + 1];
  }

  // ---- GEMM 2: out = gate @ Wb + ob ----
  float* orow = out + ((size_t)b * N_ + rowbase) * OD_;
  for (int ct = 0; ct < 4; ++ct) {
    v8f c = {};
    for (int kk = 0; kk < 16; ++kk) {
      const int k0 = kk * 4 + half * 2;
      v2f bf;
      bf.x = sWb[(k0    ) * OD_ + ct * 16 + l16];
      bf.y = sWb[(k0 + 1) * OD_ + ct * 16 + l16];
      c = __builtin_amdgcn_wmma_f32_16x16x4_f32(false, afrag[kk], false, bf,
                                                (short)0, c, false, false);
    }
    const int nn = ct * 16 + l16;
    const float bias = sOb[nn];
    for (int j = 0; j < 8; ++j) {
      const int m = j + 8 * half;
      orow[(size_t)m * OD_ + nn] = c[j] + bias;
    }
  }
}

extern "C" void kernel_launch(void* const* d_in, const int* in_sizes, int n_in,
                              void* d_out, int out_size, void* d_ws, size_t ws_size,
                              hipStream_t stream) {
  const float* qd  = (const float*)d_in[0];   // q_data
  const float* md  = (const float*)d_in[1];   // m_data
  const float* qm  = (const float*)d_in[2];   // q_mask
  // d_in[3] = bias (ignored by the forward, as in the reference)
  const float* qw  = (const float*)d_in[4];   // q_weights
  const float* kw  = (const float*)d_in[5];   // k_weights
  const float* vw  = (const float*)d_in[6];   // v_weights
  const float* ow  = (const float*)d_in[7];   // o_weights
  const float* ob  = (const float*)d_in[8];   // o_bias
  const float* gwp = (const float*)d_in[9];   // gating_w
  const float* gbp = (const float*)d_in[10];  // gating_b
  float* out = (float*)d_out;
  float* wa  = (float*)d_ws;                  // [B,64] scratch

  ga_pool_attn<<<dim3(B_), dim3(256), 0, stream>>>(qd, md, qm, qw, kw, vw, wa);
  ga_gate_out<<<dim3(B_, N_ / 64), dim3(128), 0, stream>>>(qd, gwp, gbp, ow, ob, wa, out);
}